// BatchedFidelity_47897475285428
// MI455X (gfx1250) — compile-verified
//
#include <hip/hip_runtime.h>
#include <math.h>

// ---------------------------------------------------------------------------
// Batched quantum fidelity F(A,B) = sqrt( Tr( sqrt( sqrt(A) B sqrt(A) ) ) )
// BATCH=512 density matrices, DIM=256, f32.
//
// Newton-Schulz coupled iteration for matrix sqrt -> everything is
// 256x256x256 f32 GEMMs on the CDNA5 WMMA pipe (V_WMMA_F32_16X16X4_F32).
// Frobenius scale factors fold in analytically at the end.
//
// GEMM micro-architecture (per workgroup = 128 thr = 4 wave32):
//   - 128x64 output block, wave owns 32x64 = 2 Mtiles x 4 Ntiles (64 acc VGPRs)
//   - K staged in 32-wide LDS panels; A tile padded to stride 34 (bank spread)
//   - B tile stored K-pair-interleaved so each 4x16 B fragment is ONE aligned
//     ds_load_b64 (no repack movs in front of the WMMAs)
//   - float4 (b128) global staging + global_prefetch of next panel
//
// Workspace: 5 batched 256x256 f32 matrices + per-batch scalars (~640 MB).
// ---------------------------------------------------------------------------

#define DIM     256
#define BATCH   512
#define MSTRIDE (DIM * DIM)   // 65536 elements per matrix
#define NS_ITERS 18

#define TA_3I 1   // A -> (3I - A)
#define TB_3I 2   // B -> (3I - B)

#define LA_STRIDE 34          // padded A tile row stride (floats)

typedef __attribute__((ext_vector_type(2))) float v2f;
typedef __attribute__((ext_vector_type(8))) float v8f;

// ---------------------------------------------------------------------------
// C[b] = alpha * op(A[b]) @ op(B[b]),  op(X) = X or (3I - X).
// grid = (BATCH, 8): blockIdx.y -> 2x4 grid of 128x64 blocks.
// ---------------------------------------------------------------------------
__global__ __launch_bounds__(128)
void fid_gemm(const float* __restrict__ Abase, const float* __restrict__ Bbase,
              float* __restrict__ Cbase, int flags, float alpha)
{
    __shared__ float lA[128][LA_STRIDE];  // [m][k], k-panel of 32, padded
    __shared__ float lB[16 * 128];        // K-pair interleaved: [(k>>1)][n*2+(k&1)]

    const int b  = blockIdx.x;
    const int bm = (blockIdx.y >> 2) * 128;   // 2 M-blocks
    const int bn = (blockIdx.y & 3) * 64;     // 4 N-blocks

    const float* A = Abase + (size_t)b * MSTRIDE;
    const float* B = Bbase + (size_t)b * MSTRIDE;
    float*       C = Cbase + (size_t)b * MSTRIDE;

    const int t    = threadIdx.x;
    const int wave = t >> 5;        // 0..3
    const int lane = t & 31;
    const int half = lane >> 4;     // 0: lanes 0-15, 1: lanes 16-31
    const int lm   = lane & 15;

    v8f acc[2][4];
#pragma unroll
    for (int mt = 0; mt < 2; ++mt)
#pragma unroll
        for (int tn = 0; tn < 4; ++tn) acc[mt][tn] = (v8f){0,0,0,0,0,0,0,0};

    for (int ks = 0; ks < DIM; ks += 32) {
        // ---- stage A block [bm,bm+128) x [ks,ks+32) via b128 loads ----
#pragma unroll
        for (int j = 0; j < 8; ++j) {
            int f   = t + j * 128;          // 0..1023 float4 slots
            int row = f >> 3;               // 8 float4 per 32-wide row
            int c4  = f & 7;
            int gm  = bm + row;
            const float4 v = *(const float4*)&A[gm * DIM + ks + c4 * 4];
            float e0 = v.x, e1 = v.y, e2 = v.z, e3 = v.w;
            if (flags & TA_3I) {
                int gk = ks + c4 * 4;
                e0 = (gm == gk + 0) ? (3.0f - e0) : (-e0);
                e1 = (gm == gk + 1) ? (3.0f - e1) : (-e1);
                e2 = (gm == gk + 2) ? (3.0f - e2) : (-e2);
                e3 = (gm == gk + 3) ? (3.0f - e3) : (-e3);
            }
            lA[row][c4 * 4 + 0] = e0;
            lA[row][c4 * 4 + 1] = e1;
            lA[row][c4 * 4 + 2] = e2;
            lA[row][c4 * 4 + 3] = e3;
        }
        // ---- stage B block [ks,ks+32) x [bn,bn+64), K-pair interleaved ----
#pragma unroll
        for (int j = 0; j < 4; ++j) {
            int f  = t + j * 128;           // 0..511 float4 slots
            int k  = f >> 4;                // 16 float4 per 64-wide row
            int n4 = f & 15;
            int gk = ks + k;
            const float4 v = *(const float4*)&B[gk * DIM + bn + n4 * 4];
            float e[4] = {v.x, v.y, v.z, v.w};
#pragma unroll
            for (int q = 0; q < 4; ++q) {
                int n  = n4 * 4 + q;
                float x = e[q];
                if (flags & TB_3I) x = (gk == bn + n) ? (3.0f - x) : (-x);
                lB[(k >> 1) * 128 + n * 2 + (k & 1)] = x;
            }
        }
        __syncthreads();

        // ---- prefetch next panel into caches while we compute ----
        if (ks + 32 < DIM) {
            __builtin_prefetch(&A[(bm + t) * DIM + ks + 32], 0, 1);
            if (t < 32) __builtin_prefetch(&B[(ks + 32 + t) * DIM + bn], 0, 1);
        }

        // ---- 8 sub-steps of K=4 ----
#pragma unroll
        for (int kk = 0; kk < 8; ++kk) {
            const int k0 = kk * 4 + half * 2;   // even
            // B fragments: single aligned b64 each
            v2f bf[4];
#pragma unroll
            for (int tn = 0; tn < 4; ++tn)
                bf[tn] = *(const v2f*)&lB[(k0 >> 1) * 128 + (tn * 16 + lm) * 2];
            // A fragments: contiguous pair in padded row
            v2f af[2];
#pragma unroll
            for (int mt = 0; mt < 2; ++mt)
                af[mt] = *(const v2f*)&lA[wave * 32 + mt * 16 + lm][k0];
#pragma unroll
            for (int mt = 0; mt < 2; ++mt)
#pragma unroll
                for (int tn = 0; tn < 4; ++tn)
                    acc[mt][tn] = __builtin_amdgcn_wmma_f32_16x16x4_f32(
                        false, af[mt], false, bf[tn],
                        (short)0, acc[mt][tn], false, false);
        }
        __syncthreads();
    }

    // ---- store: C/D layout, VGPR r -> M = r + 8*half ----
#pragma unroll
    for (int mt = 0; mt < 2; ++mt) {
        const int row0 = bm + wave * 32 + mt * 16 + half * 8;
        const int col0 = bn + lm;
#pragma unroll
        for (int tn = 0; tn < 4; ++tn)
#pragma unroll
            for (int r = 0; r < 8; ++r)
                C[(row0 + r) * DIM + col0 + tn * 16] = alpha * acc[mt][tn][r];
    }
}

// ---------------------------------------------------------------------------
// Frobenius norm per batch: norms[b] = ||X[b]||_F
// ---------------------------------------------------------------------------
__global__ __launch_bounds__(256)
void fid_frob(const float* __restrict__ Xbase, float* __restrict__ norms)
{
    __shared__ float red[256];
    const int b = blockIdx.x;
    const float* X = Xbase + (size_t)b * MSTRIDE;
    float s = 0.0f;
    for (int i = threadIdx.x; i < MSTRIDE; i += 256) {
        float v = X[i];
        s += v * v;
    }
    red[threadIdx.x] = s;
    __syncthreads();
    for (int off = 128; off > 0; off >>= 1) {
        if ((int)threadIdx.x < off) red[threadIdx.x] += red[threadIdx.x + off];
        __syncthreads();
    }
    if (threadIdx.x == 0) norms[b] = sqrtf(fmaxf(red[0], 1e-30f));
}

// ---------------------------------------------------------------------------
// Y[b] = X[b] / norms[b]; Z[b] = I   (Newton-Schulz init)
// ---------------------------------------------------------------------------
__global__ __launch_bounds__(256)
void fid_init(const float* __restrict__ Xbase, const float* __restrict__ norms,
              float* __restrict__ Ybase, float* __restrict__ Zbase)
{
    const int b = blockIdx.x;
    const float inv = 1.0f / norms[b];
    const float* X = Xbase + (size_t)b * MSTRIDE;
    float* Y = Ybase + (size_t)b * MSTRIDE;
    float* Z = Zbase + (size_t)b * MSTRIDE;
    for (int i = threadIdx.x; i < MSTRIDE; i += 256) {
        Y[i] = X[i] * inv;
        int r = i >> 8, c = i & 255;
        Z[i] = (r == c) ? 1.0f : 0.0f;
    }
}

// ---------------------------------------------------------------------------
// out[b] = sqrt( sqrt(c1[b]*c2[b]) * trace(Y[b]) )
// ---------------------------------------------------------------------------
__global__ __launch_bounds__(256)
void fid_finalize(const float* __restrict__ Ybase, const float* __restrict__ c1,
                  const float* __restrict__ c2, float* __restrict__ out)
{
    __shared__ float red[256];
    const int b = blockIdx.x;
    const float* Y = Ybase + (size_t)b * MSTRIDE;
    red[threadIdx.x] = Y[threadIdx.x * (DIM + 1)];   // diagonal element
    __syncthreads();
    for (int off = 128; off > 0; off >>= 1) {
        if ((int)threadIdx.x < off) red[threadIdx.x] += red[threadIdx.x + off];
        __syncthreads();
    }
    if (threadIdx.x == 0) {
        float tr    = red[0];
        float scale = sqrtf(fmaxf(c1[b] * c2[b], 0.0f));
        out[b] = sqrtf(fmaxf(scale * tr, 0.0f));
    }
}

// ---------------------------------------------------------------------------
extern "C" void kernel_launch(void* const* d_in, const int* in_sizes, int n_in,
                              void* d_out, int out_size, void* d_ws, size_t ws_size,
                              hipStream_t stream)
{
    (void)in_sizes; (void)n_in; (void)out_size; (void)ws_size;

    const float* A  = (const float*)d_in[0];
    const float* B  = (const float*)d_in[1];
    float*       out = (float*)d_out;
    float*       ws  = (float*)d_ws;

    const size_t mb = (size_t)BATCH * MSTRIDE;
    float* Y  = ws + 0 * mb;
    float* Z  = ws + 1 * mb;
    float* W  = ws + 2 * mb;
    float* Y2 = ws + 3 * mb;
    float* Z2 = ws + 4 * mb;
    float* c1 = ws + 5 * mb;
    float* c2 = c1 + BATCH;

    const dim3 gg(BATCH, 8);    // 8 blocks of 128x64 per 256x256 matrix
    const dim3 gb(128);

    // ---- phase 1: Y -> sqrt(A)/sqrt(c1) via Newton-Schulz ----
    fid_frob<<<BATCH, 256, 0, stream>>>(A, c1);
    fid_init<<<BATCH, 256, 0, stream>>>(A, c1, Y, Z);
    for (int it = 0; it < NS_ITERS; ++it) {
        fid_gemm<<<gg, gb, 0, stream>>>(Z, Y, W, 0, 1.0f);           // W  = Z@Y
        fid_gemm<<<gg, gb, 0, stream>>>(Y, W, Y2, TB_3I, 0.5f);      // Y2 = 0.5*Y@(3I-W)
        fid_gemm<<<gg, gb, 0, stream>>>(W, Z, Z2, TA_3I, 0.5f);      // Z2 = 0.5*(3I-W)@Z
        float* tmp;
        tmp = Y; Y = Y2; Y2 = tmp;
        tmp = Z; Z = Z2; Z2 = tmp;
    }

    // ---- T' = Y @ B @ Y  (actual T = c1 * T') ----
    fid_gemm<<<gg, gb, 0, stream>>>(Y, B, W, 0, 1.0f);               // W = Y@B
    fid_gemm<<<gg, gb, 0, stream>>>(W, Y, Z, 0, 1.0f);               // Z = T'

    // ---- phase 2: Newton-Schulz on T'/c2; trace of its sqrt ----
    fid_frob<<<BATCH, 256, 0, stream>>>(Z, c2);
    fid_init<<<BATCH, 256, 0, stream>>>(Z, c2, Y, W);                // Y = T'/c2, W = I
    float* Zc = W;
    float* Wc = Z;
    for (int it = 0; it < NS_ITERS; ++it) {
        fid_gemm<<<gg, gb, 0, stream>>>(Zc, Y, Wc, 0, 1.0f);
        fid_gemm<<<gg, gb, 0, stream>>>(Y, Wc, Y2, TB_3I, 0.5f);
        fid_gemm<<<gg, gb, 0, stream>>>(Wc, Zc, Z2, TA_3I, 0.5f);
        float* tmp;
        tmp = Y;  Y  = Y2; Y2 = tmp;
        tmp = Zc; Zc = Z2; Z2 = tmp;
    }

    // ---- out[b] = sqrt( sqrt(c1*c2) * Tr(Y) ) ----
    fid_finalize<<<BATCH, 256, 0, stream>>>(Y, c1, c2, out);
}